// GraphConvolutionalNetwork_34033320854157
// MI455X (gfx1250) — compile-verified
//
#include <hip/hip_runtime.h>
#include <math.h>

// ---------------------------------------------------------------------------
// Problem constants (match reference)
// ---------------------------------------------------------------------------
#define GN 100000      // nodes
#define GE 1600000     // edges / snapshot
#define GS 4           // snapshots
#define GD 128         // node dim
#define GH 256         // hidden dim
#define GP 10000       // post nodes

typedef float v2f __attribute__((ext_vector_type(2)));
typedef float v8f __attribute__((ext_vector_type(8)));

// ---------------------------------------------------------------------------
// Dense GEMM:  C[M, Nc] = X[M, K] @ W[K, Nc]   (Nc == 256)
// One block = 128 threads = 4 waves. Block b owns rows [16b, 16b+16).
// Wave w owns columns [64w, 64w+64) as four 16x16 WMMA tiles, reusing the
// A fragment across the 4 tiles per k-step.  fp32 WMMA (16x16x4) keeps full
// reference precision; GEMM is <5% of runtime so bf16 buys nothing here.
//
// Fragment layouts per CDNA5 ISA 7.12.2:
//   A (16x4):  lane l: M = l%16, K = 2*(l/16) + comp
//   B (4x16):  lane l: N = l%16, K = 2*(l/16) + comp
//   C/D 16x16: reg r:  M = r + 8*(l/16), N = l%16
// ---------------------------------------------------------------------------
__global__ void __launch_bounds__(128)
wmma_gemm_f32(const float* __restrict__ X, const float* __restrict__ W,
              float* __restrict__ C, int K, int Nc) {
    const int lane = threadIdx.x & 31;
    const int wave = threadIdx.x >> 5;       // 0..3
    const int row0 = blockIdx.x * 16;
    const int col0 = wave * 64;
    const int m    = lane & 15;
    const int kh   = (lane >> 4) * 2;        // 0 or 2

    v8f acc0 = {}, acc1 = {}, acc2 = {}, acc3 = {};

    for (int k = 0; k < K; k += 4) {
        // A fragment: two consecutive K values for row (row0+m)
        const float* ap = X + (size_t)(row0 + m) * K + (k + kh);
        v2f a; a.x = ap[0]; a.y = ap[1];

        const float* bbase = W + (size_t)(k + kh) * Nc + col0 + m;
        v2f b0; b0.x = bbase[0];       b0.y = bbase[Nc];
        v2f b1; b1.x = bbase[16];      b1.y = bbase[Nc + 16];
        v2f b2; b2.x = bbase[32];      b2.y = bbase[Nc + 32];
        v2f b3; b3.x = bbase[48];      b3.y = bbase[Nc + 48];

        acc0 = __builtin_amdgcn_wmma_f32_16x16x4_f32(false, a, false, b0, (short)0, acc0, false, false);
        acc1 = __builtin_amdgcn_wmma_f32_16x16x4_f32(false, a, false, b1, (short)0, acc1, false, false);
        acc2 = __builtin_amdgcn_wmma_f32_16x16x4_f32(false, a, false, b2, (short)0, acc2, false, false);
        acc3 = __builtin_amdgcn_wmma_f32_16x16x4_f32(false, a, false, b3, (short)0, acc3, false, false);
    }

    const int rbase = row0 + ((lane >> 4) << 3);   // +8 row offset for upper half-wave
    const int cl    = lane & 15;
    float* cb = C + (size_t)rbase * Nc + col0 + cl;
#pragma unroll
    for (int r = 0; r < 8; ++r) {
        float* cr = cb + (size_t)r * Nc;
        cr[0]  = acc0[r];
        cr[16] = acc1[r];
        cr[32] = acc2[r];
        cr[48] = acc3[r];
    }
}

// ---------------------------------------------------------------------------
// Degree: deg[dst] += 1 over edges (deg pre-zeroed; self-loop added in rsqrt)
// ---------------------------------------------------------------------------
__global__ void count_deg(const int* __restrict__ dst, float* __restrict__ deg, int E) {
    int e = blockIdx.x * blockDim.x + threadIdx.x;
    if (e < E) atomicAdd(&deg[dst[e]], 1.0f);
}

__global__ void make_dinv(float* __restrict__ deg, int n) {
    int i = blockIdx.x * blockDim.x + threadIdx.x;
    if (i < n) deg[i] = rsqrtf(1.0f + deg[i]);   // in-degree + self-loop
}

// ---------------------------------------------------------------------------
// Edge scatter:  out[dst] += msg[src] * dinv[src]*dinv[dst]
// 64 threads per edge, 4 columns (float4 gather) per thread.
// Gathers hit L2 (msg buffer = 102.4 MB < 192 MB L2); adds are f32 L2 atomics.
// ---------------------------------------------------------------------------
__global__ void __launch_bounds__(256)
scatter_edges(const float* __restrict__ msg, float* __restrict__ out,
              const int* __restrict__ src, const int* __restrict__ dst,
              const float* __restrict__ dinv, int E) {
    int tid = blockIdx.x * blockDim.x + threadIdx.x;   // E*64 = 102.4M < 2^31
    if (tid >= E * 64) return;
    int e = tid >> 6;
    int q = (tid & 63) << 2;                 // column base (0,4,...,252)
    int s = src[e], d = dst[e];
    float nrm = dinv[s] * dinv[d];
    const float4 v = *reinterpret_cast<const float4*>(msg + (size_t)s * GH + q);
    float* o = out + (size_t)d * GH + q;
    atomicAdd(o + 0, v.x * nrm);
    atomicAdd(o + 1, v.y * nrm);
    atomicAdd(o + 2, v.z * nrm);
    atomicAdd(o + 3, v.w * nrm);
}

// ---------------------------------------------------------------------------
// Layer-0 epilogue: agg = relu(agg + xw*dinv^2 + b1)   (in-place on agg)
// ---------------------------------------------------------------------------
__global__ void combine_relu(float* __restrict__ agg, const float* __restrict__ xw,
                             const float* __restrict__ dinv, const float* __restrict__ bias) {
    int idx = blockIdx.x * blockDim.x + threadIdx.x;   // N*H = 25.6M
    int i = idx >> 8, c = idx & 255;
    float di = dinv[i];
    float v = agg[idx] + xw[idx] * di * di + bias[c];
    agg[idx] = v > 0.0f ? v : 0.0f;
}

// ---------------------------------------------------------------------------
// Layer-1 epilogue into snapshot accumulator:
//   accum += hw2*dinv^2 + b2 + h   (scatter part already added atomically)
// ---------------------------------------------------------------------------
__global__ void combine_accum(float* __restrict__ accum, const float* __restrict__ hw2,
                              const float* __restrict__ h, const float* __restrict__ dinv,
                              const float* __restrict__ bias) {
    int idx = blockIdx.x * blockDim.x + threadIdx.x;
    int i = idx >> 8, c = idx & 255;
    float di = dinv[i];
    accum[idx] += hw2[idx] * di * di + bias[c] + h[idx];
}

// ---------------------------------------------------------------------------
// Classifier: 1 block (128 threads) per post node.
//   post = accum[idx]/S ; hid = relu(post@Wc1 + bc1) ; sigmoid(hid@Wc2 + bc2)
// ---------------------------------------------------------------------------
__global__ void __launch_bounds__(128)
classifier(const float* __restrict__ accum, const int* __restrict__ post_idx,
           const float* __restrict__ Wc1, const float* __restrict__ bc1,
           const float* __restrict__ Wc2, const float* __restrict__ bc2,
           float* __restrict__ out, int P) {
    __shared__ float post[GH];
    __shared__ float red[128];
    int p = blockIdx.x;
    if (p >= P) return;
    int t = threadIdx.x;
    int node = post_idx[p];
    const float inv_s = 1.0f / (float)GS;
    post[t]       = accum[(size_t)node * GH + t]       * inv_s;
    post[t + 128] = accum[(size_t)node * GH + t + 128] * inv_s;
    __syncthreads();

    float acc = bc1[t];
#pragma unroll 8
    for (int k = 0; k < GH; ++k) acc += post[k] * Wc1[k * 128 + t];
    float hid = acc > 0.0f ? acc : 0.0f;
    red[t] = hid * Wc2[t];
    __syncthreads();
    for (int s = 64; s > 0; s >>= 1) {
        if (t < s) red[t] += red[t + s];
        __syncthreads();
    }
    if (t == 0) out[p] = 1.0f / (1.0f + __expf(-(red[0] + bc2[0])));
}

// ---------------------------------------------------------------------------
// Launch: xw1 once; per snapshot: deg -> dinv -> scatter(xw1)->relu ->
// WMMA h@W2 -> scatter(hw2) into accum -> epilogue; then classifier.
// Workspace (floats): xw1 | h/agg0 | hw2 | accum | deg   (~410 MB)
// ---------------------------------------------------------------------------
extern "C" void kernel_launch(void* const* d_in, const int* in_sizes, int n_in,
                              void* d_out, int out_size, void* d_ws, size_t ws_size,
                              hipStream_t stream) {
    const float* x        = (const float*)d_in[0];
    const int*   edge     = (const int*)d_in[1];    // [S,2,E]
    const int*   post_idx = (const int*)d_in[2];
    const float* W1  = (const float*)d_in[3];
    const float* b1  = (const float*)d_in[4];
    const float* W2  = (const float*)d_in[5];
    const float* b2  = (const float*)d_in[6];
    const float* Wc1 = (const float*)d_in[7];
    const float* bc1 = (const float*)d_in[8];
    const float* Wc2 = (const float*)d_in[9];
    const float* bc2 = (const float*)d_in[10];
    float* out = (float*)d_out;

    const size_t NH = (size_t)GN * GH;          // 25.6M floats
    float* ws    = (float*)d_ws;
    float* xw1   = ws;                          // [N,H]  x@W1 (snapshot-invariant)
    float* h     = ws + NH;                     // [N,H]  layer-0 agg / h
    float* hw2   = ws + 2 * NH;                 // [N,H]  h@W2
    float* accum = ws + 3 * NH;                 // [N,H]  snapshot sum
    float* deg   = ws + 4 * NH;                 // [N]    degree -> dinv

    const int elemBlocks = (int)(NH / 256);             // 100000
    const int scatBlocks = (int)(((long long)GE * 64 + 255) / 256);  // 400000

    // xw1 = x @ W1  (K=128)
    wmma_gemm_f32<<<GN / 16, 128, 0, stream>>>(x, W1, xw1, GD, GH);
    hipMemsetAsync(accum, 0, NH * sizeof(float), stream);

    for (int s = 0; s < GS; ++s) {
        const int* src = edge + (size_t)s * 2 * GE;
        const int* dst = src + GE;

        hipMemsetAsync(deg, 0, (size_t)GN * sizeof(float), stream);
        hipMemsetAsync(h, 0, NH * sizeof(float), stream);
        count_deg<<<(GE + 255) / 256, 256, 0, stream>>>(dst, deg, GE);
        make_dinv<<<(GN + 255) / 256, 256, 0, stream>>>(deg, GN);

        // layer 0: scatter xw1 -> h, then self-loop + bias + relu
        scatter_edges<<<scatBlocks, 256, 0, stream>>>(xw1, h, src, dst, deg, GE);
        combine_relu<<<elemBlocks, 256, 0, stream>>>(h, xw1, deg, b1);

        // layer 1: hw2 = h @ W2  (K=256)
        wmma_gemm_f32<<<GN / 16, 128, 0, stream>>>(h, W2, hw2, GH, GH);

        // scatter hw2 straight into snapshot accumulator (all terms commute),
        // then add self-loop + bias + residual h
        scatter_edges<<<scatBlocks, 256, 0, stream>>>(hw2, accum, src, dst, deg, GE);
        combine_accum<<<elemBlocks, 256, 0, stream>>>(accum, hw2, h, deg, b2);
    }

    classifier<<<GP, 128, 0, stream>>>(accum, post_idx, Wc1, bc1, Wc2, bc2, out, GP);
}